// NetworkInNetwork_18030272708840
// MI455X (gfx1250) — compile-verified
//
#include <hip/hip_runtime.h>
#include <stdint.h>

typedef float v2f __attribute__((ext_vector_type(2)));
typedef float v8f __attribute__((ext_vector_type(8)));

#define D_IN  128
#define D_OUT 128
#define D_CAT 256
#define GN_EPS 1e-5f

// ---------------------------------------------------------------- zero scratch
__global__ void gnn_zero_kernel(float* __restrict__ p, long n) {
    long i = (long)blockIdx.x * blockDim.x + threadIdx.x;
    long stride = (long)gridDim.x * blockDim.x;
    for (; i < n; i += stride) p[i] = 0.0f;
}

// ---------------- repack W[K][128] -> Wp[K/2][128][2] (K-pair interleave)
// so a WMMA B-fragment (rows ka, ka+1 at column n) is one contiguous b64 load
__global__ void gnn_repack_kernel(const float* __restrict__ W, float* __restrict__ Wp, int K) {
    int idx = blockIdx.x * blockDim.x + threadIdx.x;          // over K*128
    if (idx >= K * D_OUT) return;
    int k = idx >> 7, n = idx & 127;
    Wp[(((k >> 1) * D_OUT) + n) * 2 + (k & 1)] = W[idx];
}

// ------------------------------------------------- h = ELU(x1 @ W1 + b1), WMMA
__global__ __launch_bounds__(256)
void gnn_gemm1_elu_kernel(const float* __restrict__ x1, const v2f* __restrict__ W1p,
                          const float* __restrict__ b1, float* __restrict__ h, int nNodes) {
    __shared__ float xt[16 * 130];              // 16 rows x 128 cols, padded stride
    const int row0 = blockIdx.x * 16;
    const int tid  = threadIdx.x;

    for (int i = 0; i < 8; ++i) {               // 2048 elements, 256 threads
        int idx = tid + 256 * i;
        int r = idx >> 7, c = idx & 127;
        int grow = row0 + r;
        xt[r * 130 + c] = (grow < nNodes) ? x1[(long)grow * D_IN + c] : 0.0f;
    }
    __syncthreads();

    const int wave = tid >> 5, lane = tid & 31;
    const int lanelo = lane & 15, half = lane >> 4;
    const int n0 = wave * 16;                   // 8 waves cover all 128 output cols
    const int col = n0 + lanelo;

    v8f acc = {};
    for (int k = 0; k < D_IN; k += 4) {
        int ka = k + 2 * half;                  // A: M=lanelo, K=ka..ka+1 (ISA 16x4 layout)
        v2f a;
        a.x = xt[lanelo * 130 + ka];
        a.y = xt[lanelo * 130 + ka + 1];
        v2f b = W1p[((k >> 1) + half) * D_OUT + col];   // rows {ka, ka+1}, one b64
        acc = __builtin_amdgcn_wmma_f32_16x16x4_f32(false, a, false, b,
                                                    (short)0, acc, false, false);
    }

    const float bias = b1[col];
    for (int r = 0; r < 8; ++r) {               // C: M = r + 8*half, N = lanelo
        int grow = row0 + r + 8 * half;
        if (grow < nNodes) {
            float v = acc[r] + bias;
            v = v > 0.0f ? v : expm1f(v);       // ELU(alpha=1)
            h[(long)grow * D_OUT + col] = v;
        }
    }
}

// -------------------------------------------- agg[row] += w * h[col] (1 wave/edge)
__global__ __launch_bounds__(256)
void gnn_scatter_kernel(const long long* __restrict__ ei, const float* __restrict__ ew,
                        const float* __restrict__ h, float* __restrict__ agg, long nEdges) {
    long e = (long)blockIdx.x * 8 + (threadIdx.x >> 5);
    if (e >= nEdges) return;
    const int lane = threadIdx.x & 31;
    const long row = ei[e];
    const long col = ei[nEdges + e];
    const float w  = ew[e];
    float4 v = ((const float4*)(h + col * D_OUT))[lane];   // 4 channels per lane
    float* dst = agg + row * D_OUT + (long)lane * 4;
    unsafeAtomicAdd(dst + 0, w * v.x);                     // native global_atomic_add_f32
    unsafeAtomicAdd(dst + 1, w * v.y);
    unsafeAtomicAdd(dst + 2, w * v.z);
    unsafeAtomicAdd(dst + 3, w * v.w);
}

// ------------------------------------- per-channel sum / sum-of-squares reduction
__global__ __launch_bounds__(256)
void gnn_reduce_kernel(const float* __restrict__ agg, float* __restrict__ S1,
                       float* __restrict__ S2, int nNodes) {
    __shared__ float r1[256], r2[256];
    const int c  = threadIdx.x & 127;
    const int rg = threadIdx.x >> 7;
    float s1 = 0.0f, s2 = 0.0f;
    for (long r = (long)blockIdx.x * 2 + rg; r < nNodes; r += (long)gridDim.x * 2) {
        float v = agg[r * D_OUT + c];
        s1 += v;
        s2 += v * v;
    }
    r1[threadIdx.x] = s1; r2[threadIdx.x] = s2;
    __syncthreads();
    if (rg == 0) {
        s1 += r1[threadIdx.x + 128];
        s2 += r2[threadIdx.x + 128];
        unsafeAtomicAdd(&S1[c], s1);
        unsafeAtomicAdd(&S2[c], s2);
    }
}

// -------- fold GraphNorm into per-channel affine: normed = alpha*agg + beta
__global__ void gnn_params_kernel(const float* __restrict__ S1, const float* __restrict__ S2,
                                  const float* __restrict__ gw, const float* __restrict__ gb,
                                  const float* __restrict__ gms,
                                  float* __restrict__ alpha, float* __restrict__ beta, int nNodes) {
    int c = threadIdx.x;
    if (c >= D_OUT) return;
    float invN = 1.0f / (float)nNodes;
    float mean = S1[c] * invN;
    float m2   = S2[c] * invN;
    float s    = gms[c];
    // E[(x - s*m)^2] = E[x^2] - 2*s*m^2 + s^2*m^2
    float var  = m2 - 2.0f * s * mean * mean + s * s * mean * mean;
    float rstd = rsqrtf(var + GN_EPS);
    float a    = gw[c] * rstd;
    alpha[c] = a;
    beta[c]  = gb[c] - s * mean * a;
}

// ----------------- out = [alpha*agg+beta, x1] @ W2 + b2, WMMA over K=256
__global__ __launch_bounds__(256)
void gnn_gemm2_kernel(const float* __restrict__ agg, const float* __restrict__ x1,
                      const float* __restrict__ alpha, const float* __restrict__ beta,
                      const v2f* __restrict__ W2p, const float* __restrict__ b2,
                      float* __restrict__ out, int nNodes) {
    __shared__ float at[16 * 258];              // 16 rows x 256 cols, padded stride
    const int row0 = blockIdx.x * 16;
    const int tid  = threadIdx.x;

    for (int i = 0; i < 16; ++i) {              // 4096 elements, 256 threads
        int idx = tid + 256 * i;
        int r = idx >> 8, c = idx & 255;
        int grow = row0 + r;
        float v = 0.0f;
        if (grow < nNodes) {
            if (c < 128) v = alpha[c] * agg[(long)grow * D_OUT + c] + beta[c];
            else         v = x1[(long)grow * D_IN + (c - 128)];
        }
        at[r * 258 + c] = v;
    }
    __syncthreads();

    const int wave = tid >> 5, lane = tid & 31;
    const int lanelo = lane & 15, half = lane >> 4;
    const int n0 = wave * 16;
    const int col = n0 + lanelo;

    v8f acc = {};
    for (int k = 0; k < D_CAT; k += 4) {
        int ka = k + 2 * half;
        v2f a;
        a.x = at[lanelo * 258 + ka];
        a.y = at[lanelo * 258 + ka + 1];
        v2f b = W2p[((k >> 1) + half) * D_OUT + col];
        acc = __builtin_amdgcn_wmma_f32_16x16x4_f32(false, a, false, b,
                                                    (short)0, acc, false, false);
    }

    const float bias = b2[col];
    for (int r = 0; r < 8; ++r) {
        int grow = row0 + r + 8 * half;
        if (grow < nNodes) out[(long)grow * D_OUT + col] = acc[r] + bias;
    }
}

// --------------------------------------------------------------------- launcher
extern "C" void kernel_launch(void* const* d_in, const int* in_sizes, int n_in,
                              void* d_out, int out_size, void* d_ws, size_t ws_size,
                              hipStream_t stream) {
    const float*     x1  = (const float*)d_in[0];
    const long long* ei  = (const long long*)d_in[1];   // int64 edge_index [2, E]
    const float*     ew  = (const float*)d_in[2];
    const float*     W1  = (const float*)d_in[3];
    const float*     b1  = (const float*)d_in[4];
    const float*     W2  = (const float*)d_in[5];
    const float*     b2  = (const float*)d_in[6];
    const float*     gw  = (const float*)d_in[7];
    const float*     gb  = (const float*)d_in[8];
    const float*     gms = (const float*)d_in[9];

    const int  nNodes = in_sizes[0] / D_IN;     // 100000
    const long nEdges = (long)in_sizes[2];      // 3200000

    // ws layout: agg [nNodes*128] | S1[128] | S2[128] | alpha[128] | beta[128]
    //            | W1p [128*128] | W2p [256*128]
    float* agg   = (float*)d_ws;
    float* S1    = agg + (long)nNodes * D_OUT;
    float* S2    = S1 + D_OUT;
    float* alpha = S2 + D_OUT;
    float* beta  = alpha + D_OUT;
    float* W1p   = beta + D_OUT;
    float* W2p   = W1p + D_IN * D_OUT;

    // h is staged in d_out (same size as final output), dead before gemm2 writes
    float* h   = (float*)d_out;
    float* out = (float*)d_out;

    const long nZero = (long)nNodes * D_OUT + 2 * D_OUT;   // agg + S1 + S2 (contiguous)
    gnn_zero_kernel<<<2048, 256, 0, stream>>>(agg, nZero);

    gnn_repack_kernel<<<(D_IN  * D_OUT + 255) / 256, 256, 0, stream>>>(W1, W1p, D_IN);
    gnn_repack_kernel<<<(D_CAT * D_OUT + 255) / 256, 256, 0, stream>>>(W2, W2p, D_CAT);

    const int rowTiles = (nNodes + 15) / 16;
    gnn_gemm1_elu_kernel<<<rowTiles, 256, 0, stream>>>(x1, (const v2f*)W1p, b1, h, nNodes);

    const long scatterBlocks = (nEdges + 7) / 8;
    gnn_scatter_kernel<<<(int)scatterBlocks, 256, 0, stream>>>(ei, ew, h, agg, nEdges);

    gnn_reduce_kernel<<<1024, 256, 0, stream>>>(agg, S1, S2, nNodes);
    gnn_params_kernel<<<1, 128, 0, stream>>>(S1, S2, gw, gb, gms, alpha, beta, nNodes);

    gnn_gemm2_kernel<<<rowTiles, 256, 0, stream>>>(agg, x1, alpha, beta,
                                                   (const v2f*)W2p, b2, out, nNodes);
}